// CoAttentionReadout_72782515798220
// MI455X (gfx1250) — compile-verified
//
#include <hip/hip_runtime.h>
#include <hip/hip_bf16.h>

// CoAttentionReadout for MI455X (gfx1250), fp32 via V_WMMA_F32_16X16X4_F32.
// B=16, H=8, T_TIME=64, F=128, MAX_USERS=992, L=2, dh=16.
// A-tiles staged to LDS with GLOBAL_LOAD_ASYNC_TO_LDS_B128 (ASYNCcnt path).

#define B_  16
#define H_  8
#define TT_ 64
#define F_  128
#define MU_ 992
#define L_  2
#define DH_ 16

typedef float v2f __attribute__((ext_vector_type(2)));
typedef float v8f __attribute__((ext_vector_type(8)));

__device__ __forceinline__ v8f wmma4(v2f a, v2f b, v8f c) {
  // (neg_a, A, neg_b, B, c_mod, C, reuse_a, reuse_b)
  return __builtin_amdgcn_wmma_f32_16x16x4_f32(false, a, false, b, (short)0, c,
                                               false, false);
}

__device__ __forceinline__ float wsum(float v) {
#pragma unroll
  for (int off = 16; off > 0; off >>= 1) v += __shfl_xor(v, off, 32);
  return v;
}

// Stage a 16xCOLS fp32 tile (LDS row stride STRIDE floats) using the CDNA5
// async LDS-load path: per-lane GLOBAL_LOAD_ASYNC_TO_LDS_B128, tracked by
// ASYNCcnt, fenced with s_wait_asynccnt before the workgroup barrier.
template <int COLS, int STRIDE>
__device__ __forceinline__ void async_tile16(float* sA, const float* __restrict__ A,
                                             long row0) {
  constexpr int CH = COLS / 4;  // 16-byte chunks per row
  for (int idx = threadIdx.x; idx < 16 * CH; idx += 256) {
    const int r = idx / CH;
    const int c4 = (idx % CH) * 4;
    unsigned lds = (unsigned)(uintptr_t)(sA + r * STRIDE + c4);
    unsigned long long g =
        (unsigned long long)(uintptr_t)(A + (row0 + r) * (long)COLS + c4);
    asm volatile("global_load_async_to_lds_b128 %0, %1, off"
                 :
                 : "v"(lds), "v"(g)
                 : "memory");
  }
  asm volatile("s_wait_asynccnt 0" ::: "memory");
  __syncthreads();
}

// ---------------------------------------------------------------- setup ----
__global__ void k_zero(float* __restrict__ p, long n) {
  long i = (long)blockIdx.x * blockDim.x + threadIdx.x;
  long stride = (long)gridDim.x * blockDim.x;
  for (; i < n; i += stride) p[i] = 0.0f;
}

__global__ void k_offsets(const int* __restrict__ bi, int* __restrict__ offs, int N) {
  int i = blockIdx.x * blockDim.x + threadIdx.x;
  if (i >= N) return;
  if (i == 0) offs[bi[0]] = 0;
  else if (bi[i] != bi[i - 1]) offs[bi[i]] = i;
}

__global__ void k_counts(const int* __restrict__ offs, int* __restrict__ counts, int N) {
  int t = threadIdx.x;
  if (t < B_) counts[t] = ((t == B_ - 1) ? N : offs[t + 1]) - offs[t];
}

__global__ void k_scatter(const float* __restrict__ xu, const int* __restrict__ bi,
                          const int* __restrict__ offs, float* __restrict__ padded) {
  int i = blockIdx.x;
  int f = threadIdx.x;  // 128 threads
  int b = bi[i];
  int pos = i - offs[b];
  padded[((long)b * MU_ + pos) * F_ + f] = xu[(long)i * F_ + f];
}

// ------------------------------------------------------------ projections ---
// Out(b,h,t,d) = sum_k A[row,k]*W[k,n] + bias[n], row = b*T + t, n = h*16 + d
// T is compile-time (992 or 64) so batch/time decomposition is shifts, not
// the scalar-division microcode sequence.
template <int T>
__global__ void __launch_bounds__(256)
k_gemm_qkv(const float* __restrict__ A, const float* __restrict__ W,
           const float* __restrict__ bias, float* __restrict__ Out) {
  __shared__ float sA[16 * 132];
  const int row0 = blockIdx.x * 16;
  async_tile16<F_, 132>(sA, A, row0);

  const int wave = threadIdx.x >> 5;
  const int lane = threadIdx.x & 31;
  const int hi = lane >> 4, ln = lane & 15;
  const int n = wave * 16 + ln;
  v8f acc = {};
#pragma unroll 4
  for (int kb = 0; kb < 32; ++kb) {
    const int k = kb * 4 + 2 * hi;
    v2f a = *(const v2f*)(sA + ln * 132 + k);
    v2f b;
    b.x = W[k * F_ + n];
    b.y = W[(k + 1) * F_ + n];
    acc = wmma4(a, b, acc);
  }
  const int h = wave;  // F=128 -> exactly 8 head tiles
  const int b_ = row0 / T;
  const int t0 = row0 % T;
  const float bz = bias[n];
#pragma unroll
  for (int r = 0; r < 8; ++r) {
    const int t = t0 + r + 8 * hi;
    Out[(((long)b_ * H_ + h) * T + t) * DH_ + ln] = acc[r] + bz;
  }
}

// Out[m,n] = sum_k A[m,k]*W[k,n] + bias[n] + R[m,n]   (pre-LayerNorm tensor)
__global__ void __launch_bounds__(256)
k_gemm_out(const float* __restrict__ A, const float* __restrict__ W,
           const float* __restrict__ bias, const float* __restrict__ R,
           float* __restrict__ Out) {
  __shared__ float sA[16 * 132];
  const int row0 = blockIdx.x * 16;
  async_tile16<F_, 132>(sA, A, row0);

  const int wave = threadIdx.x >> 5;
  const int lane = threadIdx.x & 31;
  const int hi = lane >> 4, ln = lane & 15;
  const int n = wave * 16 + ln;
  v8f acc = {};
#pragma unroll 4
  for (int kb = 0; kb < 32; ++kb) {
    const int k = kb * 4 + 2 * hi;
    v2f a = *(const v2f*)(sA + ln * 132 + k);
    v2f b;
    b.x = W[k * F_ + n];
    b.y = W[(k + 1) * F_ + n];
    acc = wmma4(a, b, acc);
  }
  const float bz = bias[n];
#pragma unroll
  for (int r = 0; r < 8; ++r) {
    const long m = row0 + r + 8 * hi;
    Out[m * F_ + n] = acc[r] + bz + R[m * F_ + n];
  }
}

// ----------------------------------------------------------------- LN ------
__global__ void __launch_bounds__(256)
k_ln(const float* __restrict__ X, float* __restrict__ Y,
     const float* __restrict__ gb, int M) {
  const int row = blockIdx.x * 8 + (threadIdx.x >> 5);
  const int lane = threadIdx.x & 31;
  if (row >= M) return;
  const float* x = X + (long)row * F_;
  float v[4];
  float s = 0.f;
#pragma unroll
  for (int i = 0; i < 4; ++i) { v[i] = x[lane + 32 * i]; s += v[i]; }
  s = wsum(s);
  const float mu = s * (1.0f / F_);
  float q = 0.f;
#pragma unroll
  for (int i = 0; i < 4; ++i) { float d = v[i] - mu; q += d * d; }
  q = wsum(q);
  const float rs = rsqrtf(q * (1.0f / F_) + 1e-5f);
  float* y = Y + (long)row * F_;
#pragma unroll
  for (int i = 0; i < 4; ++i) {
    int c = lane + 32 * i;
    y[c] = (v[i] - mu) * rs * gb[c] + gb[F_ + c];
  }
}

// -------------------------------------------------------------- attention --
// One wave per (b, h, 16-row q-tile): S row (992 keys) lives in LDS,
// two-pass softmax, 1/sum folded into the P@V epilogue.
__global__ void __launch_bounds__(32)
k_attn(const float* __restrict__ Q, const float* __restrict__ K,
       const float* __restrict__ V, const int* __restrict__ counts,
       float* __restrict__ O, int TQ) {
  __shared__ float sS[16 * MU_];
  __shared__ float sinv[16];
  const int qt = blockIdx.x, h = blockIdx.y, b = blockIdx.z;
  const int lane = threadIdx.x;
  const int hi = lane >> 4, ln = lane & 15;
  const int cnt = counts[b];
  const float* Qp = Q + (((long)b * H_ + h) * TQ + qt * 16) * DH_;
  const float* Kp = K + ((long)b * H_ + h) * (long)MU_ * DH_;
  const float* Vp = V + ((long)b * H_ + h) * (long)MU_ * DH_;

  v2f qf[4];
#pragma unroll
  for (int kb = 0; kb < 4; ++kb)
    qf[kb] = *(const v2f*)(Qp + ln * DH_ + kb * 4 + 2 * hi);

  // S = (Q K^T) * 1/sqrt(16), masked -> LDS
  for (int jt = 0; jt < MU_ / 16; ++jt) {
    v8f acc = {};
    const float* Kr = Kp + (jt * 16 + ln) * DH_;
#pragma unroll
    for (int kb = 0; kb < 4; ++kb) {
      v2f bf = *(const v2f*)(Kr + kb * 4 + 2 * hi);
      acc = wmma4(qf[kb], bf, acc);
    }
    const int j = jt * 16 + ln;
    const bool masked = (j >= cnt);
#pragma unroll
    for (int r = 0; r < 8; ++r) {
      float s = acc[r] * 0.25f;
      s = masked ? -1.0e9f : s;
      sS[(r + 8 * hi) * MU_ + j] = s;
    }
  }

  // per-row max / expsum (2 lanes per row)
  const int row = lane >> 1, par = lane & 1;
  float mx = -3.0e38f;
  for (int j = par; j < MU_; j += 2) mx = fmaxf(mx, sS[row * MU_ + j]);
  mx = fmaxf(mx, __shfl_xor(mx, 1, 32));
  float sum = 0.f;
  for (int j = par; j < MU_; j += 2) {
    float e = __expf(sS[row * MU_ + j] - mx);
    sS[row * MU_ + j] = e;
    sum += e;
  }
  sum += __shfl_xor(sum, 1, 32);
  if (par == 0) sinv[row] = 1.0f / sum;

  // O = (P @ V) * sinv[row]
  v8f accO = {};
  for (int jt = 0; jt < MU_ / 16; ++jt) {
#pragma unroll
    for (int kb = 0; kb < 4; ++kb) {
      const int j = jt * 16 + kb * 4 + 2 * hi;
      v2f a = *(const v2f*)(sS + ln * MU_ + j);
      v2f bf;
      bf.x = Vp[j * DH_ + ln];
      bf.y = Vp[(j + 1) * DH_ + ln];
      accO = wmma4(a, bf, accO);
    }
  }
#pragma unroll
  for (int r = 0; r < 8; ++r) {
    const int m = r + 8 * hi;
    O[((long)b * TQ + qt * 16 + m) * F_ + h * DH_ + ln] = accO[r] * sinv[m];
  }
}

// ---------------------------------------------------------------- readout --
__global__ void k_factors(const float* __restrict__ y, const float* __restrict__ padded,
                          const float* __restrict__ x, const float* __restrict__ xt,
                          float* __restrict__ g) {
  const int b = blockIdx.x;
  const int f = threadIdx.x;  // 128 threads
  float uf = 0.f, sp = 0.f;
  for (int t = 0; t < MU_; ++t) {
    float p = padded[((long)b * MU_ + t) * F_ + f];
    uf += y[((long)b * MU_ + t) * F_ + f] + p;
    sp += p;
  }
  float tf = x[((long)b * TT_ + (TT_ - 1)) * F_ + f] +
             xt[((long)b * TT_ + (TT_ - 1)) * F_ + f];
  g[b * 512 + f] = tf;
  g[b * 512 + 128 + f] = uf;
  g[b * 512 + 256 + f] = uf * tf;
  g[b * 512 + 384 + f] = sp;
}

__global__ void __launch_bounds__(256)
k_mlp1(const float* __restrict__ G, const float* __restrict__ W1,
       const float* __restrict__ b1, float* __restrict__ Hout) {
  __shared__ float sA[16 * 516];
  async_tile16<512, 516>(sA, G, 0);

  const int wave = threadIdx.x >> 5;
  const int lane = threadIdx.x & 31;
  const int hi = lane >> 4, ln = lane & 15;
  const int n = (blockIdx.x * 8 + wave) * 16 + ln;
  v8f acc = {};
  for (int kb = 0; kb < 128; ++kb) {
    const int k = kb * 4 + 2 * hi;
    v2f a = *(const v2f*)(sA + ln * 516 + k);
    v2f b;
    b.x = W1[k * 512 + n];
    b.y = W1[(k + 1) * 512 + n];
    acc = wmma4(a, b, acc);
  }
  const float bz = b1[n];
#pragma unroll
  for (int r = 0; r < 8; ++r) Hout[(r + 8 * hi) * 512 + n] = acc[r] + bz;
}

__global__ void __launch_bounds__(512)
k_head(const float* __restrict__ h1, const float* __restrict__ mlp_ln,
       const float* __restrict__ W2, const float* __restrict__ b2,
       float* __restrict__ out) {
  const int row = threadIdx.x >> 5;  // 16 waves, one per batch row
  const int lane = threadIdx.x & 31;
  float v[16];
  float s = 0.f;
#pragma unroll
  for (int i = 0; i < 16; ++i) { v[i] = h1[row * 512 + lane + 32 * i]; s += v[i]; }
  s = wsum(s);
  const float mu = s * (1.0f / 512.0f);
  float q = 0.f;
#pragma unroll
  for (int i = 0; i < 16; ++i) { float d = v[i] - mu; q += d * d; }
  q = wsum(q);
  const float rs = rsqrtf(q * (1.0f / 512.0f) + 1e-5f);
  float dot = 0.f;
#pragma unroll
  for (int i = 0; i < 16; ++i) {
    int c = lane + 32 * i;
    float hv = (v[i] - mu) * rs * mlp_ln[c] + mlp_ln[512 + c];
    hv = fmaxf(hv, 0.f);
    dot += hv * W2[c];
  }
  dot = wsum(dot);
  if (lane == 0) {
    float z = dot + b2[0];
    out[row] = (z > 0.f) ? z : 0.5f * (__expf(2.0f * z) - 1.0f);  // celu a=0.5
  }
}

// ------------------------------------------------------------------ host ---
extern "C" void kernel_launch(void* const* d_in, const int* in_sizes, int n_in,
                              void* d_out, int out_size, void* d_ws, size_t ws_size,
                              hipStream_t stream) {
  (void)n_in; (void)out_size; (void)ws_size;
  const float* x_user = (const float*)d_in[0];
  const int*   bidx   = (const int*)d_in[1];
  const float* x_time = (const float*)d_in[2];
  const float* awy    = (const float*)d_in[3];
  const float* aby    = (const float*)d_in[4];
  const float* lnyp   = (const float*)d_in[5];
  const float* awx    = (const float*)d_in[6];
  const float* abx    = (const float*)d_in[7];
  const float* lnxp   = (const float*)d_in[8];
  const float* w1     = (const float*)d_in[9];
  const float* b1     = (const float*)d_in[10];
  const float* mlpln  = (const float*)d_in[11];
  const float* w2     = (const float*)d_in[12];
  const float* b2     = (const float*)d_in[13];
  const int N = in_sizes[0] / F_;

  const size_t P  = (size_t)B_ * MU_ * F_;  // 2,031,616 floats
  const size_t XT = (size_t)B_ * TT_ * F_;  //   131,072 floats
  float* padded = (float*)d_ws;
  float* ybuf = padded + P;
  float* xbuf = ybuf + P;
  float* Qb = xbuf + XT;
  float* Kb = Qb + P;
  float* Vb = Kb + P;
  float* Ob = Vb + P;
  float* gbuf = Ob + P;
  float* h1buf = gbuf + (size_t)B_ * 4 * F_;
  int* counts = (int*)(h1buf + (size_t)B_ * 4 * F_);
  int* offs = counts + B_;

  k_zero<<<dim3(2048), 256, 0, stream>>>(padded, (long)P);
  k_offsets<<<dim3((N + 255) / 256), 256, 0, stream>>>(bidx, offs, N);
  k_counts<<<dim3(1), 32, 0, stream>>>(offs, counts, N);
  k_scatter<<<dim3(N), 128, 0, stream>>>(x_user, bidx, offs, padded);
  hipMemcpyAsync(ybuf, padded, P * sizeof(float), hipMemcpyDeviceToDevice, stream);
  hipMemcpyAsync(xbuf, x_time, XT * sizeof(float), hipMemcpyDeviceToDevice, stream);

  const int MTY = B_ * MU_ / 16;  // 992 row-tiles
  const int MTX = B_ * TT_ / 16;  // 64 row-tiles
  for (int l = 0; l < L_; ++l) {
    const float* Wy = awy + (size_t)l * 4 * F_ * F_;
    const float* by = aby + (size_t)l * 4 * F_;
    k_gemm_qkv<MU_><<<dim3(MTY), 256, 0, stream>>>(ybuf, Wy + 0 * F_ * F_, by + 0 * F_, Qb);
    k_gemm_qkv<MU_><<<dim3(MTY), 256, 0, stream>>>(ybuf, Wy + 1 * F_ * F_, by + 1 * F_, Kb);
    k_gemm_qkv<MU_><<<dim3(MTY), 256, 0, stream>>>(ybuf, Wy + 2 * F_ * F_, by + 2 * F_, Vb);
    k_attn<<<dim3(MU_ / 16, H_, B_), 32, 0, stream>>>(Qb, Kb, Vb, counts, Ob, MU_);
    k_gemm_out<<<dim3(MTY), 256, 0, stream>>>(Ob, Wy + 3 * F_ * F_, by + 3 * F_, ybuf, Qb);
    k_ln<<<dim3(B_ * MU_ / 8), 256, 0, stream>>>(Qb, ybuf, lnyp + (size_t)l * 2 * F_, B_ * MU_);

    const float* Wx = awx + (size_t)l * 4 * F_ * F_;
    const float* bx = abx + (size_t)l * 4 * F_;
    k_gemm_qkv<TT_><<<dim3(MTX), 256, 0, stream>>>(xbuf, Wx + 0 * F_ * F_, bx + 0 * F_, Qb);
    k_gemm_qkv<MU_><<<dim3(MTY), 256, 0, stream>>>(ybuf, Wx + 1 * F_ * F_, bx + 1 * F_, Kb);
    k_gemm_qkv<MU_><<<dim3(MTY), 256, 0, stream>>>(ybuf, Wx + 2 * F_ * F_, bx + 2 * F_, Vb);
    k_attn<<<dim3(TT_ / 16, H_, B_), 32, 0, stream>>>(Qb, Kb, Vb, counts, Ob, TT_);
    k_gemm_out<<<dim3(MTX), 256, 0, stream>>>(Ob, Wx + 3 * F_ * F_, bx + 3 * F_, xbuf, Kb);
    k_ln<<<dim3(B_ * TT_ / 8), 256, 0, stream>>>(Kb, xbuf, lnxp + (size_t)l * 2 * F_, B_ * TT_);
  }

  k_factors<<<dim3(B_), 128, 0, stream>>>(ybuf, padded, xbuf, x_time, gbuf);
  k_mlp1<<<dim3(4), 256, 0, stream>>>(gbuf, w1, b1, h1buf);
  k_head<<<dim3(1), 512, 0, stream>>>(h1buf, mlpln, w2, b2, (float*)d_out);
}